// GraphSAGEOGBGPPA_49220325212352
// MI455X (gfx1250) — compile-verified
//
#include <hip/hip_runtime.h>
#include <hip/hip_bf16.h>

#define NN 100000
#define EE 1250000
#define DD 64
#define DE 7
#define LL 5
#define GG 128
#define TT 37

typedef __attribute__((ext_vector_type(2))) float v2f;
typedef __attribute__((ext_vector_type(8))) float v8f;

// ---------------------------------------------------------------------------
// zero fill
// ---------------------------------------------------------------------------
__global__ void zero_f32(float* __restrict__ p, long n) {
    long i = (long)blockIdx.x * blockDim.x + threadIdx.x;
    long stride = (long)gridDim.x * blockDim.x;
    for (; i < n; i += stride) p[i] = 0.0f;
}

// ---------------------------------------------------------------------------
// Per-layer message + scatter:
//   agg[dst] += relu(h[src] + edge_attr[e] @ W_edge + b_edge)
// One thread per edge, full D=64 row, edge_emb recomputed on the fly
// (W_edge/b_edge loads are uniform -> scalar cache).
// ---------------------------------------------------------------------------
__global__ void msg_scatter(const float* __restrict__ h,
                            const int*   __restrict__ ei,     // [2*E]
                            const float* __restrict__ eattr,  // [E,7]
                            const float* __restrict__ We,     // [7,64]
                            const float* __restrict__ be,     // [64]
                            float* __restrict__ agg) {
    int e = blockIdx.x * blockDim.x + threadIdx.x;
    if (e >= EE) return;
    int src = ei[e];
    int dst = ei[EE + e];
    float ea[DE];
#pragma unroll
    for (int k = 0; k < DE; ++k) ea[k] = eattr[(long)e * DE + k];
    const float* hs = h + (long)src * DD;
    float* ad = agg + (long)dst * DD;
#pragma unroll 4
    for (int d0 = 0; d0 < DD; d0 += 4) {
        float4 hv  = *(const float4*)(hs + d0);
        float4 emb = *(const float4*)(be + d0);
#pragma unroll
        for (int k = 0; k < DE; ++k) {
            float4 w = *(const float4*)(We + k * DD + d0);
            emb.x += ea[k] * w.x; emb.y += ea[k] * w.y;
            emb.z += ea[k] * w.z; emb.w += ea[k] * w.w;
        }
        float mx = fmaxf(hv.x + emb.x, 0.0f);
        float my = fmaxf(hv.y + emb.y, 0.0f);
        float mz = fmaxf(hv.z + emb.z, 0.0f);
        float mw = fmaxf(hv.w + emb.w, 0.0f);
        unsafeAtomicAdd(ad + d0 + 0, mx);
        unsafeAtomicAdd(ad + d0 + 1, my);
        unsafeAtomicAdd(ad + d0 + 2, mz);
        unsafeAtomicAdd(ad + d0 + 3, mw);
    }
}

// ---------------------------------------------------------------------------
// GIN MLP update via f32 WMMA:
//   h_next = relu( ((1+eps)*h + agg) @ W + b ),  also written to node_emb slice.
// One wave -> 16 rows x 64 cols of output; 16 K-steps x 4 N-tiles of
// V_WMMA_F32_16X16X4_F32. A is fused (1+eps)*h+agg, built per K-step.
// ---------------------------------------------------------------------------
__global__ void gin_update(const float* __restrict__ h,
                           const float* __restrict__ agg,
                           const float* __restrict__ W,       // [64,64] row-major (Kin x Nout)
                           const float* __restrict__ b,       // [64]
                           const float* __restrict__ eps_arr, // [L]
                           int layer,
                           float* __restrict__ h_next,
                           float* __restrict__ node_emb) {
    const int lane = threadIdx.x & 31;
    const int wave = threadIdx.x >> 5;
    const int r0   = blockIdx.x * 128 + wave * 16;   // 8 waves * 16 rows
    const int lm   = lane & 15;                       // row-in-tile / col-in-tile
    const int lh   = lane >> 4;                       // 0 or 1 (K-pair select)
    const float se = 1.0f + eps_arr[layer];

    int rowA = r0 + lm;
    if (rowA >= NN) rowA = NN - 1;                    // clamp; stores masked later
    const float* hp = h   + (long)rowA * DD;
    const float* ap = agg + (long)rowA * DD;

    v8f acc0 = {}, acc1 = {}, acc2 = {}, acc3 = {};

#pragma unroll
    for (int kt = 0; kt < 16; ++kt) {
        const int kb = kt * 4 + 2 * lh;               // this lane's K base (kb, kb+1)
        v2f a;
        a.x = se * hp[kb]     + ap[kb];
        a.y = se * hp[kb + 1] + ap[kb + 1];

        v2f b0, b1, b2, b3;
        b0.x = W[kb * DD +  0 + lm]; b0.y = W[(kb + 1) * DD +  0 + lm];
        b1.x = W[kb * DD + 16 + lm]; b1.y = W[(kb + 1) * DD + 16 + lm];
        b2.x = W[kb * DD + 32 + lm]; b2.y = W[(kb + 1) * DD + 32 + lm];
        b3.x = W[kb * DD + 48 + lm]; b3.y = W[(kb + 1) * DD + 48 + lm];

        acc0 = __builtin_amdgcn_wmma_f32_16x16x4_f32(false, a, false, b0, (short)0, acc0, false, false);
        acc1 = __builtin_amdgcn_wmma_f32_16x16x4_f32(false, a, false, b1, (short)0, acc1, false, false);
        acc2 = __builtin_amdgcn_wmma_f32_16x16x4_f32(false, a, false, b2, (short)0, acc2, false, false);
        acc3 = __builtin_amdgcn_wmma_f32_16x16x4_f32(false, a, false, b3, (short)0, acc3, false, false);
    }

    // Epilogue: bias + relu, store to h_next and node_emb[:, layer*64 : (layer+1)*64]
    const float bb0 = b[ 0 + lm], bb1 = b[16 + lm], bb2 = b[32 + lm], bb3 = b[48 + lm];
#pragma unroll
    for (int j = 0; j < 8; ++j) {
        int row = r0 + j + 8 * lh;                    // C/D layout: VGPR j -> M = j (+8 for hi lanes)
        if (row < NN) {
            float v0 = fmaxf(acc0[j] + bb0, 0.0f);
            float v1 = fmaxf(acc1[j] + bb1, 0.0f);
            float v2 = fmaxf(acc2[j] + bb2, 0.0f);
            float v3 = fmaxf(acc3[j] + bb3, 0.0f);
            float* hn = h_next + (long)row * DD;
            hn[ 0 + lm] = v0; hn[16 + lm] = v1; hn[32 + lm] = v2; hn[48 + lm] = v3;
            float* ne = node_emb + (long)row * (LL * DD) + layer * DD;
            ne[ 0 + lm] = v0; ne[16 + lm] = v1; ne[32 + lm] = v2; ne[48 + lm] = v3;
        }
    }
}

// ---------------------------------------------------------------------------
// Per-graph sum pooling (+ node counts). gsum is [G,320] (L2-resident, 160KB).
// Thread per (node, 4-col chunk): 80 chunks of 4 per node.
// ---------------------------------------------------------------------------
__global__ void pool_sum(const float* __restrict__ ne,
                         const int*   __restrict__ batch,
                         float* __restrict__ gsum,   // [G,320]
                         float* __restrict__ cnt) {  // [G]
    long t = (long)blockIdx.x * blockDim.x + threadIdx.x;
    long total = (long)NN * 80;
    if (t >= total) return;
    int n  = (int)(t / 80);
    int c4 = (int)(t % 80) * 4;
    int g  = batch[n];
    float4 v = *(const float4*)(ne + (long)n * (LL * DD) + c4);
    float* gp = gsum + (long)g * (LL * DD) + c4;
    unsafeAtomicAdd(gp + 0, v.x);
    unsafeAtomicAdd(gp + 1, v.y);
    unsafeAtomicAdd(gp + 2, v.z);
    unsafeAtomicAdd(gp + 3, v.w);
    if (c4 == 0) unsafeAtomicAdd(cnt + g, 1.0f);
}

// ---------------------------------------------------------------------------
// Readout: out[g,t] = b[t] + sum_c (gsum[g,c]/max(cnt[g],1)) * Wp[c,t]
// G*T = 4736 threads, 320 MACs each — negligible.
// ---------------------------------------------------------------------------
__global__ void readout(const float* __restrict__ gsum,
                        const float* __restrict__ cnt,
                        const float* __restrict__ Wp,  // [320,37]
                        const float* __restrict__ bp,  // [37]
                        float* __restrict__ out) {     // [G,37]
    int t = blockIdx.x * blockDim.x + threadIdx.x;
    if (t >= GG * TT) return;
    int g = t / TT, c = t % TT;
    float inv = 1.0f / fmaxf(cnt[g], 1.0f);
    float s = bp[c];
    const float* gr = gsum + (long)g * (LL * DD);
#pragma unroll 8
    for (int k = 0; k < LL * DD; ++k) s += gr[k] * inv * Wp[(long)k * TT + c];
    out[t] = s;
}

// ---------------------------------------------------------------------------
extern "C" void kernel_launch(void* const* d_in, const int* in_sizes, int n_in,
                              void* d_out, int out_size, void* d_ws, size_t ws_size,
                              hipStream_t stream) {
    const float* x     = (const float*)d_in[0];
    const int*   ei    = (const int*)  d_in[1];
    const float* eattr = (const float*)d_in[2];
    const int*   batch = (const int*)  d_in[3];
    const float* We    = (const float*)d_in[4];
    const float* be    = (const float*)d_in[5];
    const float* eps   = (const float*)d_in[6];
    const float* Wm    = (const float*)d_in[7];
    const float* bm    = (const float*)d_in[8];
    const float* Wp    = (const float*)d_in[9];
    const float* bp    = (const float*)d_in[10];
    float* out = (float*)d_out;

    // workspace carve-out
    char* ws = (char*)d_ws;
    size_t off = 0;
    float* hbuf0 = (float*)(ws + off); off += (size_t)NN * DD * sizeof(float);
    float* hbuf1 = (float*)(ws + off); off += (size_t)NN * DD * sizeof(float);
    float* agg   = (float*)(ws + off); off += (size_t)NN * DD * sizeof(float);
    float* nemb  = (float*)(ws + off); off += (size_t)NN * LL * DD * sizeof(float);
    float* gsum  = (float*)(ws + off); off += (size_t)GG * LL * DD * sizeof(float);
    float* cnt   = (float*)(ws + off); off += (size_t)GG * sizeof(float);

    const int ZB = 1024;
    const float* hcur = x;
    float* hnxt = hbuf0;

    for (int l = 0; l < LL; ++l) {
        zero_f32<<<1024, 256, 0, stream>>>(agg, (long)NN * DD);
        msg_scatter<<<(EE + 255) / 256, 256, 0, stream>>>(hcur, ei, eattr, We, be, agg);
        gin_update<<<(NN + 127) / 128, 256, 0, stream>>>(
            hcur, agg, Wm + (long)l * DD * DD, bm + (long)l * DD, eps, l, hnxt, nemb);
        hcur = hnxt;
        hnxt = (hcur == hbuf0) ? hbuf1 : hbuf0;
    }

    // gsum and cnt are contiguous: zero both in one pass
    zero_f32<<<(GG * LL * DD + GG + ZB - 1) / ZB, ZB, 0, stream>>>(gsum, (long)GG * LL * DD + GG);
    long pool_threads = (long)NN * 80;
    pool_sum<<<(int)((pool_threads + 255) / 256), 256, 0, stream>>>(nemb, batch, gsum, cnt);
    readout<<<(GG * TT + 255) / 256, 256, 0, stream>>>(gsum, cnt, Wp, bp, out);
}